// GCN_Net_48524540511071
// MI455X (gfx1250) — compile-verified
//
#include <hip/hip_runtime.h>

#define N_NODES 100000
#define F_IN    256
#define HID     128
#define NCLS    16
#define NE      1600000

typedef float v8f __attribute__((ext_vector_type(8)));
typedef float v2f __attribute__((ext_vector_type(2)));

// ---------------- degree / normalization ----------------

__global__ void deg_init(int* __restrict__ deg) {
    int i = blockIdx.x * 256 + threadIdx.x;
    if (i < N_NODES) deg[i] = 1;            // self-loop
}

__global__ void deg_count(const int* __restrict__ dst, int* __restrict__ deg) {
    int e = blockIdx.x * 256 + threadIdx.x;
    if (e < NE) atomicAdd(&deg[dst[e]], 1);
}

__global__ void dinv_k(const int* __restrict__ deg, float* __restrict__ dinv) {
    int i = blockIdx.x * 256 + threadIdx.x;
    if (i < N_NODES) dinv[i] = rsqrtf((float)deg[i]);
}

// ---------------- GEMM1: h0 = x @ W1  (fp32 WMMA 16x16x4) ----------------
// One block handles a 16-row stripe of x (staged in LDS); 8 waves cover the
// 128 output columns as 8 16-wide tiles.

__global__ void __launch_bounds__(256) gemm1_wmma(const float* __restrict__ x,
                                                  const float* __restrict__ W1,
                                                  float* __restrict__ h0) {
    __shared__ float sA[16 * F_IN];                 // 16 KB
    const int row_base = blockIdx.x * 16;

    // coalesced float4 staging: 16*256/4 = 1024 float4, 4 per thread
    {
        const float4* s4 = (const float4*)(x + (size_t)row_base * F_IN);
        float4* d4 = (float4*)sA;
        for (int i = threadIdx.x; i < 1024; i += 256) d4[i] = s4[i];
    }
    __syncthreads();

    const int wave = threadIdx.x >> 5;              // n-tile 0..7
    const int lane = threadIdx.x & 31;
    const int half = lane >> 4;                     // K split across half-waves
    const int l16  = lane & 15;
    const int col  = wave * 16 + l16;               // output column [0,128)

    v8f c = {};
    for (int k0 = 0; k0 < F_IN; k0 += 4) {
        const int kb = k0 + half * 2;
        v2f a, b;
        a.x = sA[l16 * F_IN + kb];
        a.y = sA[l16 * F_IN + kb + 1];
        b.x = W1[(size_t)kb * HID + col];
        b.y = W1[(size_t)(kb + 1) * HID + col];
        c = __builtin_amdgcn_wmma_f32_16x16x4_f32(false, a, false, b,
                                                  (short)0, c, false, false);
    }
    const int mrow = row_base + half * 8;           // C/D layout: VGPR v -> M=v(+8)
#pragma unroll
    for (int v = 0; v < 8; ++v)
        h0[(size_t)(mrow + v) * HID + col] = c[v];
}

// ---------------- aggregation 1 ----------------

// init accumulator with self-loop term: hagg = h0 * dinv^2
__global__ void agg1_init(const float* __restrict__ h0,
                          const float* __restrict__ dinv,
                          float* __restrict__ hagg) {
    int t = blockIdx.x * 256 + threadIdx.x;         // over N*HID/4 float4s
    if (t >= N_NODES * (HID / 4)) return;
    int i = t / (HID / 4);
    float w = dinv[i]; w *= w;
    float4 v = ((const float4*)h0)[t];
    v.x *= w; v.y *= w; v.z *= w; v.w *= w;
    ((float4*)hagg)[t] = v;
}

// one wave per edge: float4 gather from L2-resident h0, 4 f32 atomics per lane
__global__ void __launch_bounds__(256) agg1_edges(const int* __restrict__ src,
                                                  const int* __restrict__ dst,
                                                  const float* __restrict__ dinv,
                                                  const float* __restrict__ h0,
                                                  float* __restrict__ hagg) {
    const int e = blockIdx.x * 8 + (threadIdx.x >> 5);
    if (e >= NE) return;
    const int lane = threadIdx.x & 31;
    const int s = src[e], d = dst[e];
    const float norm = dinv[s] * dinv[d];
    const float4 v = *(const float4*)(h0 + (size_t)s * HID + lane * 4);
    float* o = hagg + (size_t)d * HID + lane * 4;
    atomicAdd(o + 0, v.x * norm);
    atomicAdd(o + 1, v.y * norm);
    atomicAdd(o + 2, v.z * norm);
    atomicAdd(o + 3, v.w * norm);
}

__global__ void bias_relu(float* __restrict__ hagg, const float* __restrict__ b1) {
    int t = blockIdx.x * 256 + threadIdx.x;
    if (t >= N_NODES * HID) return;
    float v = hagg[t] + b1[t & (HID - 1)];
    hagg[t] = v > 0.f ? v : 0.f;
}

// ---------------- GEMM2: h2 = h1 @ W2 (fp32 WMMA, N=16) ----------------

__global__ void __launch_bounds__(256) gemm2_wmma(const float* __restrict__ h1,
                                                  const float* __restrict__ W2,
                                                  float* __restrict__ h2) {
    const int wg = blockIdx.x * 8 + (threadIdx.x >> 5);
    if (wg >= N_NODES / 16) return;                 // wave-uniform guard
    const int lane = threadIdx.x & 31;
    const int half = lane >> 4;
    const int l16  = lane & 15;
    const int row_base = wg * 16;

    v8f c = {};
    for (int k0 = 0; k0 < HID; k0 += 4) {
        const int kb = k0 + half * 2;
        v2f a, b;
        a.x = h1[(size_t)(row_base + l16) * HID + kb];
        a.y = h1[(size_t)(row_base + l16) * HID + kb + 1];
        b.x = W2[kb * NCLS + l16];
        b.y = W2[(kb + 1) * NCLS + l16];
        c = __builtin_amdgcn_wmma_f32_16x16x4_f32(false, a, false, b,
                                                  (short)0, c, false, false);
    }
#pragma unroll
    for (int v = 0; v < 8; ++v)
        h2[(size_t)(row_base + half * 8 + v) * NCLS + l16] = c[v];
}

// ---------------- aggregation 2 (into d_out) ----------------

// out = h2 * dinv^2 + b2 (self-loop + bias folded into init)
__global__ void out_init(const float* __restrict__ h2,
                         const float* __restrict__ dinv,
                         const float* __restrict__ b2,
                         float* __restrict__ out) {
    int t = blockIdx.x * 256 + threadIdx.x;
    if (t >= N_NODES * NCLS) return;
    float w = dinv[t >> 4];
    out[t] = h2[t] * w * w + b2[t & 15];
}

// thread per (edge, feature); 16 consecutive lanes share one edge
__global__ void agg2_edges(const int* __restrict__ src,
                           const int* __restrict__ dst,
                           const float* __restrict__ dinv,
                           const float* __restrict__ h2,
                           float* __restrict__ out) {
    int t = blockIdx.x * 256 + threadIdx.x;
    if (t >= NE * NCLS) return;
    int e = t >> 4, f = t & 15;
    int s = src[e], d = dst[e];
    float norm = dinv[s] * dinv[d];
    atomicAdd(&out[(size_t)d * NCLS + f], h2[(size_t)s * NCLS + f] * norm);
}

// ---------------- launch ----------------

extern "C" void kernel_launch(void* const* d_in, const int* in_sizes, int n_in,
                              void* d_out, int out_size, void* d_ws, size_t ws_size,
                              hipStream_t stream) {
    const float* x  = (const float*)d_in[0];
    const int*   ei = (const int*)d_in[1];
    const float* W1 = (const float*)d_in[2];
    const float* b1 = (const float*)d_in[3];
    const float* W2 = (const float*)d_in[4];
    const float* b2 = (const float*)d_in[5];
    float* out = (float*)d_out;
    const int* src = ei;
    const int* dst = ei + NE;

    char* ws = (char*)d_ws;
    auto align = [](size_t v) { return (v + 255) & ~(size_t)255; };
    int*   deg  = (int*)ws;    ws += align((size_t)N_NODES * 4);
    float* dinv = (float*)ws;  ws += align((size_t)N_NODES * 4);
    float* h0   = (float*)ws;  ws += align((size_t)N_NODES * HID * 4);
    float* hagg = (float*)ws;  ws += align((size_t)N_NODES * HID * 4);
    float* h2   = (float*)ws;  // N*NCLS*4

    deg_init  <<<(N_NODES + 255) / 256, 256, 0, stream>>>(deg);
    deg_count <<<(NE + 255) / 256,      256, 0, stream>>>(dst, deg);
    dinv_k    <<<(N_NODES + 255) / 256, 256, 0, stream>>>(deg, dinv);

    gemm1_wmma<<<N_NODES / 16, 256, 0, stream>>>(x, W1, h0);

    agg1_init <<<(N_NODES * (HID / 4) + 255) / 256, 256, 0, stream>>>(h0, dinv, hagg);
    agg1_edges<<<(NE + 7) / 8, 256, 0, stream>>>(src, dst, dinv, h0, hagg);
    bias_relu <<<(N_NODES * HID + 255) / 256, 256, 0, stream>>>(hagg, b1);

    gemm2_wmma<<<(N_NODES / 16 + 7) / 8, 256, 0, stream>>>(hagg, W2, h2);

    out_init  <<<(N_NODES * NCLS + 255) / 256, 256, 0, stream>>>(h2, dinv, b2, out);
    agg2_edges<<<(NE * NCLS + 255) / 256, 256, 0, stream>>>(src, dst, dinv, h2, out);
}